// StandaloneMELayer_54752243090066
// MI455X (gfx1250) — compile-verified
//
#include <hip/hip_runtime.h>
#include <stdint.h>

// ---------------------------------------------------------------------------
// StandaloneMELayer scan for MI455X (gfx1250)
//   x:(8,4096,1024) f32, tape_init:(1024), eta:(1), torque_bias:(1024)
//   out:(8,4096,1024) f32
// One block per batch row; 256 threads x float4 = D=1024.
// Sequential over T. Per step:
//   - branchless update (VALU, packed)
//   - wave32 sum via DPP row_xmask + permlanex16 (pure VALU)
//   - cross-wave combine via 8-float LDS exchange + minimal split barrier
//     (s_wait_dscnt 0 ; s_barrier_signal -1 ... s_barrier_wait -1), with the
//     async refill + next-tile LDS load hidden inside the barrier window
//   - async global->LDS streaming of x (ASYNCcnt), 8 stages deep
//   - nontemporal b128 output stores
// ---------------------------------------------------------------------------

#define T_LEN    4096
#define D_LEN    1024
#define VEC      4
#define NTH      256                 // == D_LEN / VEC
#define NWAVES   (NTH / 32)
#define NSTAGES  8
#define EPS_SIGN 1e-6f
#define EPS_N2   1e-16f              // (1e-8)^2 : max(sqrt(q),1e-8) == sqrt(max(q,1e-16))

typedef float vfloat4 __attribute__((ext_vector_type(4)));

struct LdsLayout {
    vfloat4 h[NSTAGES][NTH];                               // streamed x tiles
    __attribute__((aligned(16))) float partial[2][NWAVES]; // double-buffered
};

// ---- gfx1250 async global->LDS + asynccnt wait (inline asm; CDNA5 path) ----
__device__ __forceinline__ uint32_t to_lds_off(void* p) {
#if defined(__HIP_DEVICE_COMPILE__)
    return (uint32_t)(uintptr_t)(__attribute__((address_space(3))) void*)p;
#else
    (void)p; return 0u;
#endif
}

__device__ __forceinline__ void async_load_b128(void* lds_dst, const void* gsrc) {
#if defined(__HIP_DEVICE_COMPILE__)
    uint32_t loff = to_lds_off(lds_dst);
    asm volatile("global_load_async_to_lds_b128 %0, %1, off"
                 :
                 : "v"(loff), "v"(gsrc)
                 : "memory");
#else
    (void)lds_dst; (void)gsrc;
#endif
}

template <int N>
__device__ __forceinline__ void wait_asynccnt_le() {
#if defined(__HIP_DEVICE_COMPILE__)
    asm volatile("s_wait_asynccnt %0" : : "n"(N) : "memory");
#endif
}

// Minimal split barrier: only drain this wave's LDS ops, then signal.
__device__ __forceinline__ void barrier_signal_dsonly() {
#if defined(__HIP_DEVICE_COMPILE__)
    asm volatile("s_wait_dscnt 0x0\n\t"
                 "s_barrier_signal -1" : : : "memory");
#endif
}
__device__ __forceinline__ void barrier_wait() {
#if defined(__HIP_DEVICE_COMPILE__)
    asm volatile("s_barrier_wait -1" : : : "memory");
#endif
}

// ---- wave32 all-lanes sum: pure VALU (DPP row_xmask + permlanex16) ---------
__device__ __forceinline__ float wave32_sum(float v) {
    int t;
    t = __builtin_amdgcn_update_dpp(0, __builtin_bit_cast(int, v), 0x161, 0xf, 0xf, true);
    v += __builtin_bit_cast(float, t);
    t = __builtin_amdgcn_update_dpp(0, __builtin_bit_cast(int, v), 0x162, 0xf, 0xf, true);
    v += __builtin_bit_cast(float, t);
    t = __builtin_amdgcn_update_dpp(0, __builtin_bit_cast(int, v), 0x164, 0xf, 0xf, true);
    v += __builtin_bit_cast(float, t);
    t = __builtin_amdgcn_update_dpp(0, __builtin_bit_cast(int, v), 0x168, 0xf, 0xf, true);
    v += __builtin_bit_cast(float, t);
    t = __builtin_amdgcn_permlanex16(__builtin_bit_cast(int, v),
                                     __builtin_bit_cast(int, v),
                                     0x76543210, 0xfedcba98, true, false);
    v += __builtin_bit_cast(float, t);
    return v;
}

__device__ __forceinline__ float combine_partials(const float* part) {
    const vfloat4* pv = (const vfloat4*)part;        // 8 partials = 2 x b128
    vfloat4 a = pv[0], b = pv[1];
    return ((a.x + a.y) + (a.z + a.w)) + ((b.x + b.y) + (b.z + b.w));
}

// ---- kernel ------------------------------------------------------------------
__global__ __launch_bounds__(NTH, 1)
void me_scan_kernel(const float* __restrict__ x,
                    const float* __restrict__ tape_init,
                    const float* __restrict__ eta_p,
                    const float* __restrict__ tbias,
                    float* __restrict__ out)
{
    __shared__ LdsLayout lds;

    const int tid  = threadIdx.x;
    const int lane = tid & 31;
    const int wid  = tid >> 5;
    const int b    = blockIdx.x;

    const vfloat4* xb = (const vfloat4*)(x   + (size_t)b * T_LEN * D_LEN) + tid;
    vfloat4*       ob = (vfloat4*)      (out + (size_t)b * T_LEN * D_LEN) + tid;

    const float eta = fabsf(eta_p[0]);
    const vfloat4 tb  = ((const vfloat4*)tbias)[tid];
    const vfloat4 etb = { eta * tb.x, eta * tb.y, eta * tb.z, eta * tb.w };

    // ---- s0 = renorm(tape_init) (same for every batch row) ----
    vfloat4 s = ((const vfloat4*)tape_init)[tid];
    {
        float q  = s.x * s.x + s.y * s.y + s.z * s.z + s.w * s.w;
        float ws = wave32_sum(q);
        if (lane == 0) lds.partial[0][wid] = ws;
        __syncthreads();
        float tot = combine_partials(lds.partial[0]);
        float inv = __builtin_amdgcn_rsqf(fmaxf(tot, EPS_N2));
        s.x *= inv; s.y *= inv; s.z *= inv; s.w *= inv;
        __syncthreads();   // partial[0] is reused at t=0
    }

    // ---- prologue: fill NSTAGES async stages, preload h0 ----
#pragma unroll
    for (int i = 0; i < NSTAGES; ++i)
        async_load_b128(&lds.h[i][tid], xb + (size_t)i * (D_LEN / VEC));

    wait_asynccnt_le<NSTAGES - 1>();
    vfloat4 h = lds.h[0][tid];

    // ---- sequential scan over T ----
    for (int t = 0; t < T_LEN; ++t) {
        // Branchless update. 1-orth = res+tor, so:
        //   |c|>eps  -> u += eta*c ;  c<-eps -> u += eta*torque_bias
        float cx = h.x * s.x, cy = h.y * s.y, cz = h.z * s.z, cw = h.w * s.w;

        float ux = (fabsf(cx) > EPS_SIGN) ? eta * cx : 0.f;
        float uy = (fabsf(cy) > EPS_SIGN) ? eta * cy : 0.f;
        float uz = (fabsf(cz) > EPS_SIGN) ? eta * cz : 0.f;
        float uw = (fabsf(cw) > EPS_SIGN) ? eta * cw : 0.f;

        ux += (cx < -EPS_SIGN) ? etb.x : 0.f;
        uy += (cy < -EPS_SIGN) ? etb.y : 0.f;
        uz += (cz < -EPS_SIGN) ? etb.z : 0.f;
        uw += (cw < -EPS_SIGN) ? etb.w : 0.f;

        float vx = s.x + ux, vy = s.y + uy, vz = s.z + uz, vw = s.w + uw;
        float q  = vx * vx + vy * vy + vz * vz + vw * vw;

        // Wave-level sum (pure VALU), publish per-wave partial.
        float ws = wave32_sum(q);
        if (lane == 0) lds.partial[t & 1][wid] = ws;

        // Split barrier: drain only DS, signal, and hide independent work
        // (async refill + next tile's LDS load) in the signal->wait window.
        barrier_signal_dsonly();

        if (t + NSTAGES < T_LEN)
            async_load_b128(&lds.h[t & (NSTAGES - 1)][tid],
                            xb + (size_t)(t + NSTAGES) * (D_LEN / VEC));

        vfloat4 hn = h;
        if (t + 1 < T_LEN) {
            wait_asynccnt_le<NSTAGES - 1>();
            hn = lds.h[(t + 1) & (NSTAGES - 1)][tid];
        }

        barrier_wait();

        // Cross-wave combine + renorm: inv = 1/max(sqrt(q),1e-8)
        float tot = combine_partials(lds.partial[t & 1]);
        float inv = __builtin_amdgcn_rsqf(fmaxf(tot, EPS_N2));
        s.x = vx * inv; s.y = vy * inv; s.z = vz * inv; s.w = vw * inv;

        // Streaming store of this step's state (written once, never re-read).
        __builtin_nontemporal_store(s, ob + (size_t)t * (D_LEN / VEC));

        h = hn;
    }
}

// ---------------------------------------------------------------------------
extern "C" void kernel_launch(void* const* d_in, const int* in_sizes, int n_in,
                              void* d_out, int out_size, void* d_ws, size_t ws_size,
                              hipStream_t stream) {
    (void)n_in; (void)out_size; (void)d_ws; (void)ws_size;

    const float* x    = (const float*)d_in[0];
    const float* tape = (const float*)d_in[1];
    const float* eta  = (const float*)d_in[2];
    const float* tb   = (const float*)d_in[3];
    float*       out  = (float*)d_out;

    const int B = in_sizes[0] / (T_LEN * D_LEN);   // == 8

    hipLaunchKernelGGL(me_scan_kernel, dim3(B), dim3(NTH), 0, stream,
                       x, tape, eta, tb, out);
}